// MSDeformableAttention3D_PyTorch_26989574488777
// MI455X (gfx1250) — compile-verified
//
#include <hip/hip_runtime.h>

typedef float v2f __attribute__((ext_vector_type(2)));
typedef float v8f __attribute__((ext_vector_type(8)));
typedef int   v4i __attribute__((ext_vector_type(4)));

#define NQ_E       256
#define HEADS      8
#define LEVELS     4
#define POINTS     8
#define HD         32

#define AS1 __attribute__((address_space(1)))
#define AS3 __attribute__((address_space(3)))

// --- gfx1250 async global->LDS staging helpers -----------------------------
#if defined(__has_builtin)
#if __has_builtin(__builtin_amdgcn_global_load_async_to_lds_b128)
#define HAVE_ASYNC_LDS_BUILTIN 1
#endif
#if __has_builtin(__builtin_amdgcn_s_wait_asynccnt)
#define HAVE_WAIT_ASYNC_BUILTIN 1
#endif
#endif

static __device__ __forceinline__ void async_copy_b128(const float* gsrc, float* lptr)
{
#if defined(HAVE_ASYNC_LDS_BUILTIN)
    // signature: (global v4i* src, shared v4i* dst, imm offset, imm cpol)
    __builtin_amdgcn_global_load_async_to_lds_b128(
        (AS1 v4i*)gsrc, (AS3 v4i*)lptr, 0, 0);
#else
    // VDST = per-lane LDS byte address, VADDR = 64-bit global address (GV mode)
    unsigned loff = (unsigned)(size_t)(AS3 float*)lptr;
    asm volatile("global_load_async_to_lds_b128 %0, %1, off"
                 :: "v"(loff), "v"((AS1 const float*)gsrc) : "memory");
#endif
}

static __device__ __forceinline__ void wait_asynccnt0()
{
#if defined(HAVE_WAIT_ASYNC_BUILTIN)
    __builtin_amdgcn_s_wait_asynccnt(0);
#else
    asm volatile("s_wait_asynccnt 0x0" ::: "memory");
#endif
}

// ---------------------------------------------------------------------------
// GEMM: out = X[M][K] @ W[N][K]^T + bias[N]
// block = 256 threads = 8 waves; block tile = 128(M) x 64(N); wave tile 16x64.
// B panel (64 x K) staged straight into LDS via async loads (ASYNCcnt),
// shared by all 8 waves.
// head_split==1 stores out[((n>>5)*M + m)*32 + (n&31)]  (value projection)
// head_split==0 stores out[m*N + n]                      (offsets / attn)
// ---------------------------------------------------------------------------
__global__ __launch_bounds__(256) void gemm_wmma_f32(
    const float* __restrict__ X, const float* __restrict__ W,
    const float* __restrict__ bias, float* __restrict__ out,
    int M, int N, int K, int head_split)
{
    __shared__ float Bs[64 * 260];   // 64 cols x K=256, padded stride 260

    const int lane = threadIdx.x & 31;
    const int wave = threadIdx.x >> 5;
    const int n0   = blockIdx.y * 64;
    int m0 = blockIdx.x * 128 + wave * 16;

    // Async stage of B panel: 64 rows x 256 floats = 4096 x b128 chunks,
    // global -> LDS directly (no VGPR round-trip), tracked by ASYNCcnt.
    for (int i = threadIdx.x; i < 64 * 64; i += 256) {
        const int row = i >> 6;        // 0..63 (col of output)
        const int c4  = i & 63;        // float4 index within K
        async_copy_b128(W + (size_t)(n0 + row) * K + c4 * 4,
                        &Bs[row * 260 + c4 * 4]);
    }
    wait_asynccnt0();
    __syncthreads();

    // A fragment addressing (16x4 f32 layout): lanes 0-15 -> K={0,1}, 16-31 -> K={2,3}
    const int klo  = (lane >> 4) * 2;
    int mrow = m0 + (lane & 15);
    if (mrow > M - 1) mrow = M - 1;            // clamp loads; stores guarded
    const float* xrow = X + (size_t)mrow * K + klo;
    const float* bp   = &Bs[(lane & 15) * 260 + klo];

    v8f acc[4];
#pragma unroll
    for (int j = 0; j < 4; ++j) acc[j] = v8f{0.f,0.f,0.f,0.f,0.f,0.f,0.f,0.f};

#pragma unroll 4
    for (int kk = 0; kk < K; kk += 4) {
        v2f a  = *(const v2f*)(xrow + kk);
        v2f b0 = *(const v2f*)(bp + kk);
        v2f b1 = *(const v2f*)(bp + 16 * 260 + kk);
        v2f b2 = *(const v2f*)(bp + 32 * 260 + kk);
        v2f b3 = *(const v2f*)(bp + 48 * 260 + kk);
        acc[0] = __builtin_amdgcn_wmma_f32_16x16x4_f32(false, a, false, b0, (short)0, acc[0], false, false);
        acc[1] = __builtin_amdgcn_wmma_f32_16x16x4_f32(false, a, false, b1, (short)0, acc[1], false, false);
        acc[2] = __builtin_amdgcn_wmma_f32_16x16x4_f32(false, a, false, b2, (short)0, acc[2], false, false);
        acc[3] = __builtin_amdgcn_wmma_f32_16x16x4_f32(false, a, false, b3, (short)0, acc[3], false, false);
    }

    // C/D layout: VGPR r, lanes 0-15 -> M=r, lanes 16-31 -> M=r+8; N = n-base + (lane&15)
    const int rbase = m0 + ((lane >> 4) ? 8 : 0);
#pragma unroll
    for (int j = 0; j < 4; ++j) {
        const int col = n0 + j * 16 + (lane & 15);
        const float bv = bias[col];
#pragma unroll
        for (int r = 0; r < 8; ++r) {
            const int row = rbase + r;
            if (row < M) {
                const float val = acc[j][r] + bv;
                if (head_split)
                    out[((size_t)(col >> 5) * (size_t)M + (size_t)row) * 32 + (col & 31)] = val;
                else
                    out[(size_t)row * N + col] = val;
            }
        }
    }
}

// ---------------------------------------------------------------------------
// Softmax over 32 contiguous logits per (query, head); one wave per row.
// ---------------------------------------------------------------------------
__global__ __launch_bounds__(256) void softmax32_kernel(float* __restrict__ attn, int nrows)
{
    const int row = blockIdx.x * 8 + (threadIdx.x >> 5);
    if (row >= nrows) return;
    const int lane = threadIdx.x & 31;
    float x = attn[(size_t)row * 32 + lane];
    float m = x;
#pragma unroll
    for (int o = 16; o > 0; o >>= 1) m = fmaxf(m, __shfl_xor(m, o, 32));
    float e = __expf(x - m);
    float s = e;
#pragma unroll
    for (int o = 16; o > 0; o >>= 1) s += __shfl_xor(s, o, 32);
    attn[(size_t)row * 32 + lane] = e / s;
}

// ---------------------------------------------------------------------------
// Deformable sampling: one wave per (query, head); lane = channel (HD=32).
// vproj layout: [head][pixel][32ch] -> each bilinear corner = 128B coalesced.
// ---------------------------------------------------------------------------
__global__ __launch_bounds__(256) void msda_sample_kernel(
    const float* __restrict__ vproj, const float* __restrict__ off,
    const float* __restrict__ attn,  const float* __restrict__ refpt,
    float* __restrict__ out, int nq, int nv)
{
    const int q    = blockIdx.x;
    const int h    = threadIdx.x >> 5;
    const int lane = threadIdx.x & 31;

    // per-lane preload of the 32 (level,point) scalars for this (q,h)
    const float2 off_l = ((const float2*)off)[(size_t)q * 256 + h * 32 + lane];
    const float  a_l   = attn[(size_t)q * 256 + h * 32 + lane];
    const float  rp    = (lane < 8) ? refpt[(size_t)q * 8 + lane] : 0.0f;

    const int Hs[LEVELS]     = {116, 58, 29, 15};
    const int Ws[LEVELS]     = {200, 100, 50, 25};
    const int starts[LEVELS] = {0, 23200, 29000, 30450};

    float acc = 0.0f;
#pragma unroll
    for (int lvl = 0; lvl < LEVELS; ++lvl) {
        const int Hh = Hs[lvl], Ww = Ws[lvl];
        const float fW = (float)Ww, fH = (float)Hh;
        const float* vb = vproj + ((size_t)h * (size_t)nv + (size_t)starts[lvl]) * 32 + lane;
#pragma unroll
        for (int p = 0; p < POINTS; ++p) {
            const int s  = lvl * POINTS + p;
            const float ox = __shfl(off_l.x, s, 32);
            const float oy = __shfl(off_l.y, s, 32);
            const float aw = __shfl(a_l, s, 32);
            const int z = p & 3;                       // p = pa*nZ + z
            const float rx = __shfl(rp, 2 * z, 32);
            const float ry = __shfl(rp, 2 * z + 1, 32);
            // grid_sample(align_corners=False) with grid = 2*loc-1 collapses to:
            const float x = fmaf(rx, fW, ox) - 0.5f;
            const float y = fmaf(ry, fH, oy) - 0.5f;
            const float x0f = floorf(x), y0f = floorf(y);
            const int xi = (int)x0f, yi = (int)y0f;
            const float wx1 = x - x0f, wy1 = y - y0f;
            const float wx0 = 1.0f - wx1, wy0 = 1.0f - wy1;
            const bool vx0 = (xi >= 0) && (xi < Ww);
            const bool vx1 = (xi + 1 >= 0) && (xi + 1 < Ww);
            const bool vy0 = (yi >= 0) && (yi < Hh);
            const bool vy1 = (yi + 1 >= 0) && (yi + 1 < Hh);
            float v00 = 0.f, v10 = 0.f, v01 = 0.f, v11 = 0.f;
            if (vy0 && vx0) v00 = vb[(yi * Ww + xi) * 32];
            if (vy0 && vx1) v10 = vb[(yi * Ww + xi + 1) * 32];
            if (vy1 && vx0) v01 = vb[((yi + 1) * Ww + xi) * 32];
            if (vy1 && vx1) v11 = vb[((yi + 1) * Ww + xi + 1) * 32];
            const float bil = v00 * (wx0 * wy0) + v10 * (wx1 * wy0)
                            + v01 * (wx0 * wy1) + v11 * (wx1 * wy1);
            acc = fmaf(aw, bil, acc);
        }
    }
    // output E index = h*HD + hd
    out[(size_t)q * NQ_E + h * HD + lane] = acc;
}

// ---------------------------------------------------------------------------
extern "C" void kernel_launch(void* const* d_in, const int* in_sizes, int n_in,
                              void* d_out, int out_size, void* d_ws, size_t ws_size,
                              hipStream_t stream)
{
    const float* query  = (const float*)d_in[0];
    const float* value  = (const float*)d_in[1];
    const float* refpt  = (const float*)d_in[2];
    const float* W_val  = (const float*)d_in[3];
    const float* b_val  = (const float*)d_in[4];
    const float* W_off  = (const float*)d_in[5];
    const float* b_off  = (const float*)d_in[6];
    const float* W_attn = (const float*)d_in[7];
    const float* b_attn = (const float*)d_in[8];
    float* out = (float*)d_out;

    const int nq = in_sizes[0] / 256;   // 40000
    const int nv = in_sizes[1] / 256;   // 30825
    const int Noff = HEADS * LEVELS * POINTS * 2;  // 512
    const int Natt = HEADS * LEVELS * POINTS;      // 256

    float* vproj = (float*)d_ws;                          // nv * 256
    float* off   = vproj + (size_t)nv * 256;              // nq * 512
    float* attn  = off   + (size_t)nq * Noff;             // nq * 256

    dim3 blk(256);
    // 1) value projection -> head-split layout [head][pixel][32]
    gemm_wmma_f32<<<dim3((nv + 127) / 128, 256 / 64), blk, 0, stream>>>(
        value, W_val, b_val, vproj, nv, 256, 256, 1);
    // 2) sampling offsets
    gemm_wmma_f32<<<dim3((nq + 127) / 128, Noff / 64), blk, 0, stream>>>(
        query, W_off, b_off, off, nq, Noff, 256, 0);
    // 3) attention logits
    gemm_wmma_f32<<<dim3((nq + 127) / 128, Natt / 64), blk, 0, stream>>>(
        query, W_attn, b_attn, attn, nq, Natt, 256, 0);
    // 4) softmax over 32 per (q, head)
    softmax32_kernel<<<dim3((nq * HEADS + 7) / 8), blk, 0, stream>>>(attn, nq * HEADS);
    // 5) bilinear deformable sampling + weighted accumulate
    msda_sample_kernel<<<dim3(nq), blk, 0, stream>>>(vproj, off, attn, refpt, out, nq, nv);
}